// FilterDetections_42709154791983
// MI455X (gfx1250) — compile-verified
//
#include <hip/hip_runtime.h>
#include <stdint.h>

#define BDIM     1024
#define NANCH    100000
#define NCLS     30
#define NB       16
#define TOPK     200           // PRE_NMS_TOPK
#define MAXDET   100
#define CHUNK    4096
#define NBINS    16384         // float-bits >> 16, scores < 2.0
#define CANDCAP  2048
#define FLATN    (NCLS * TOPK) // 6000
#define SORTN    8192
#define SCORE_TH 0.01f
#define IOU_TH   0.5f

typedef unsigned int v4u __attribute__((ext_vector_type(4)));
typedef int          v8i __attribute__((ext_vector_type(8)));
typedef int          v4i __attribute__((ext_vector_type(4)));

// ---- Tensor Data Mover: DMA a strided column (stride elems of 4B) into LDS ----
// D# per CDNA5 ISA: 2D tensor, tile = 1 x tile_rows,
// tensor_dim0=1, tensor_dim1=rows (OOB rows read as zero), data_size=4B.
__device__ __forceinline__ void tdm_issue_col(unsigned lds_off, const float* gptr,
                                              unsigned rows, unsigned tile_rows,
                                              unsigned stride_elems) {
  unsigned long long ga = (unsigned long long)(uintptr_t)gptr;
  v4u g0;
  g0[0] = 1u;                                              // count=1, user descriptor
  g0[1] = lds_off;                                         // lds_addr (bytes)
  g0[2] = (unsigned)ga;                                    // global_addr[31:0]
  g0[3] = ((unsigned)(ga >> 32) & 0x01FFFFFFu) | (2u << 30); // addr[56:32] | type=2
  v8i g1;
  g1[0] = (int)(2u << 16);                                 // data_size=4B
  g1[1] = (int)(1u << 16);                                 // tensor_dim0 = 1
  g1[2] = (int)((rows & 0xFFFFu) << 16);                   // tensor_dim1[15:0]
  g1[3] = (int)(((rows >> 16) & 0xFFFFu) | (1u << 16));    // tensor_dim1[31:16] | tile_dim0=1
  g1[4] = (int)(tile_rows & 0xFFFFu);                      // tile_dim1 (tile_dim2=0)
  g1[5] = (int)stride_elems;                               // tensor_dim0_stride[31:0]
  g1[6] = 0;
  g1[7] = 0;
  v4i gz = {0, 0, 0, 0};
#if defined(__clang_major__) && __clang_major__ >= 23
  v8i gz8 = {0, 0, 0, 0, 0, 0, 0, 0};
  __builtin_amdgcn_tensor_load_to_lds(g0, g1, gz, gz, gz8, 0);
#else
  __builtin_amdgcn_tensor_load_to_lds(g0, g1, gz, gz, 0);
#endif
}

struct __align__(16) SmemA {
  union {
    unsigned int       hist[NBINS];     // 64 KB (pass 1)
    unsigned long long cand[CANDCAP];   // 16 KB (pass 2, aliased)
  } u;
  float buf[2][CHUNK];                  // 32 KB double-buffered stream tiles
  unsigned int seg[BDIM];               // 4 KB segment sums
  float nx1[TOPK], ny1[TOPK], nx2[TOPK], ny2[TOPK];
  int   keep[TOPK];
  unsigned int cnt;
  unsigned int thrBin;
};

__global__ __launch_bounds__(BDIM) void fd42709_topk_nms(
    const float* __restrict__ boxes, const float* __restrict__ cls,
    unsigned int* __restrict__ ws_hi, unsigned int* __restrict__ ws_anchor) {
  __shared__ SmemA s;
  const int tid = threadIdx.x;
  const int bI  = blockIdx.x / NCLS;
  const int cI  = blockIdx.x % NCLS;
  const float* colBase = cls + (size_t)bI * NANCH * NCLS + cI;
  const int nChunks = (NANCH + CHUNK - 1) / CHUNK;
  const bool wave0 = (tid < 32);
  // Generic shared-pointer low 32 bits == LDS byte offset (aperture in high bits).
  const unsigned lds0 = (unsigned)(uintptr_t)&s.buf[0][0];
  const unsigned lds1 = (unsigned)(uintptr_t)&s.buf[1][0];

  for (int i = tid; i < NBINS; i += BDIM) s.u.hist[i] = 0u;
  if (tid == 0) s.cnt = 0u;
  __syncthreads();

  // ---------- pass 1: TDM-streamed histogram of score float-bits ----------
  if (wave0) tdm_issue_col(lds0, colBase, NANCH, CHUNK, NCLS);
  for (int k = 0; k < nChunks; ++k) {
    if (wave0) {
      if (k + 1 < nChunks) {
        tdm_issue_col((k & 1) ? lds0 : lds1, colBase + (size_t)(k + 1) * CHUNK * NCLS,
                      (unsigned)(NANCH - (k + 1) * CHUNK), CHUNK, NCLS);
        __builtin_amdgcn_s_wait_tensorcnt(1);
      } else {
        __builtin_amdgcn_s_wait_tensorcnt(0);
      }
    }
    __syncthreads();
    const float* bp = s.buf[k & 1];
    const int n0 = k * CHUNK;
    for (int t = tid; t < CHUNK; t += BDIM) {
      const int n = n0 + t;
      const float v = bp[t];
      if (n < NANCH && v > SCORE_TH) {
        unsigned bin = __float_as_uint(v) >> 16;
        if (bin > NBINS - 1) bin = NBINS - 1;
        atomicAdd(&s.u.hist[bin], 1u);
      }
    }
    __syncthreads();
  }

  // ---------- find threshold bin T: largest bin with suffix-count >= TOPK ----------
  unsigned segsum = 0;
  for (int q = 0; q < NBINS / BDIM; ++q) segsum += s.u.hist[tid * (NBINS / BDIM) + q];
  s.seg[tid] = segsum;
  __syncthreads();
  if (tid == 0) {
    unsigned acc = 0;
    int sg = BDIM - 1;
    for (; sg >= 0; --sg) {
      const unsigned c = s.seg[sg];
      if (acc + c >= TOPK) break;
      acc += c;
    }
    unsigned T = 0;
    if (sg >= 0) {
      const int lo = sg * (NBINS / BDIM);
      int b2 = lo + (NBINS / BDIM) - 1;
      for (; b2 > lo; --b2) {
        const unsigned h = s.u.hist[b2];
        if (acc + h >= TOPK) break;
        acc += h;
      }
      T = (unsigned)b2;
    }
    s.thrBin = T;
  }
  __syncthreads();
  const unsigned T = s.thrBin;

  // ---------- pass 2: TDM-streamed collect of candidates with bin >= T ----------
  if (wave0) tdm_issue_col(lds0, colBase, NANCH, CHUNK, NCLS);
  for (int k = 0; k < nChunks; ++k) {
    if (wave0) {
      if (k + 1 < nChunks) {
        tdm_issue_col((k & 1) ? lds0 : lds1, colBase + (size_t)(k + 1) * CHUNK * NCLS,
                      (unsigned)(NANCH - (k + 1) * CHUNK), CHUNK, NCLS);
        __builtin_amdgcn_s_wait_tensorcnt(1);
      } else {
        __builtin_amdgcn_s_wait_tensorcnt(0);
      }
    }
    __syncthreads();
    const float* bp = s.buf[k & 1];
    const int n0 = k * CHUNK;
    for (int t = tid; t < CHUNK; t += BDIM) {
      const int n = n0 + t;
      const float v = bp[t];
      if (n < NANCH && v > SCORE_TH) {
        const unsigned bits = __float_as_uint(v);
        unsigned bin = bits >> 16;
        if (bin > NBINS - 1) bin = NBINS - 1;
        if (bin >= T) {
          const unsigned pos = atomicAdd(&s.cnt, 1u);
          if (pos < CANDCAP)  // key: score-bits desc, then index asc (via ~n)
            s.u.cand[pos] = ((unsigned long long)bits << 32) | (unsigned)(~(unsigned)n);
        }
      }
    }
    __syncthreads();
  }
  unsigned total = s.cnt;
  if (total > CANDCAP) total = CANDCAP;
  for (int i = tid; i < CANDCAP; i += BDIM)
    if ((unsigned)i >= total) s.u.cand[i] = 0ull;
  __syncthreads();

  // ---------- bitonic sort (descending) of 2048 candidate keys ----------
  for (int kk = 2; kk <= CANDCAP; kk <<= 1) {
    for (int jj = kk >> 1; jj > 0; jj >>= 1) {
      for (int i = tid; i < CANDCAP; i += BDIM) {
        const int ix = i ^ jj;
        if (ix > i) {
          const unsigned long long a = s.u.cand[i], b = s.u.cand[ix];
          const bool desc = ((i & kk) == 0);
          if (desc ? (a < b) : (a > b)) { s.u.cand[i] = b; s.u.cand[ix] = a; }
        }
      }
      __syncthreads();
    }
  }

  // ---------- gather top-200 boxes ----------
  if (tid < TOPK) {
    const unsigned long long key = s.u.cand[tid];
    const unsigned hi = (unsigned)(key >> 32);
    const unsigned anchor = ~(unsigned)key;
    const int valid = (hi != 0u);
    float4 bb = make_float4(0.f, 0.f, 0.f, 0.f);
    if (valid) bb = *(const float4*)(boxes + ((size_t)bI * NANCH + anchor) * 4);
    s.nx1[tid] = bb.x; s.ny1[tid] = bb.y; s.nx2[tid] = bb.z; s.ny2[tid] = bb.w;
    s.keep[tid] = valid;
  }
  __syncthreads();

  // ---------- greedy NMS on a single wave32 ----------
  // Cross-lane keep[] visibility: volatile forces LDS reloads; hardware keeps
  // per-wave DS ops in order, so no barriers are required inside one wave.
  if (wave0) {
    volatile int* kp = s.keep;
    for (int i = 0; i < TOPK - 1; ++i) {
      if (kp[i] != 0) {
        const float ax1 = s.nx1[i], ay1 = s.ny1[i], ax2 = s.nx2[i], ay2 = s.ny2[i];
        const float aArea = fmaxf(ax2 - ax1, 0.f) * fmaxf(ay2 - ay1, 0.f);
        for (int j = i + 1 + tid; j < TOPK; j += 32) {
          if (kp[j] != 0) {
            const float bx1 = s.nx1[j], by1 = s.ny1[j], bx2 = s.nx2[j], by2 = s.ny2[j];
            const float bArea = fmaxf(bx2 - bx1, 0.f) * fmaxf(by2 - by1, 0.f);
            const float ix1 = fmaxf(ax1, bx1), iy1 = fmaxf(ay1, by1);
            const float ix2 = fminf(ax2, bx2), iy2 = fminf(ay2, by2);
            const float inter = fmaxf(ix2 - ix1, 0.f) * fmaxf(iy2 - iy1, 0.f);
            const float uni = fmaxf(aArea + bArea - inter, 1e-8f);
            if (inter > IOU_TH * uni) kp[j] = 0;
          }
        }
      }
    }
  }
  __syncthreads();

  // ---------- write per-class survivors ----------
  if (tid < TOPK) {
    const unsigned long long key = s.u.cand[tid];
    const unsigned hi = (unsigned)(key >> 32);
    const unsigned anchor = ~(unsigned)key;
    const size_t o = ((size_t)bI * NCLS + cI) * TOPK + tid;
    ws_hi[o] = (s.keep[tid] != 0) ? hi : 0u;   // 0 == -inf / invalid
    ws_anchor[o] = anchor;
  }
}

__global__ __launch_bounds__(BDIM) void fd42709_final_topk(
    const float* __restrict__ boxes, const float* __restrict__ rot,
    const float* __restrict__ tr, const unsigned int* __restrict__ ws_hi,
    const unsigned int* __restrict__ ws_anchor, float* __restrict__ out) {
  __shared__ unsigned long long keys[SORTN];   // 64 KB
  const int tid = threadIdx.x;
  const int bI = blockIdx.x;
  const unsigned int* hiRow = ws_hi + (size_t)bI * FLATN;

  for (int i = tid; i < SORTN; i += BDIM) {
    unsigned long long key = 0ull;
    if (i < FLATN) {
      const unsigned hi = hiRow[i];
      if (hi) key = ((unsigned long long)hi << 32) | (unsigned)(~(unsigned)i);
    }
    keys[i] = key;
  }
  __syncthreads();

  for (int kk = 2; kk <= SORTN; kk <<= 1) {
    for (int jj = kk >> 1; jj > 0; jj >>= 1) {
      for (int i = tid; i < SORTN; i += BDIM) {
        const int ix = i ^ jj;
        if (ix > i) {
          const unsigned long long a = keys[i], b = keys[ix];
          const bool desc = ((i & kk) == 0);
          if (desc ? (a < b) : (a > b)) { keys[i] = b; keys[ix] = a; }
        }
      }
      __syncthreads();
    }
  }

  if (tid < MAXDET) {
    const unsigned long long key = keys[tid];
    const unsigned hi = (unsigned)(key >> 32);
    float* outBoxes  = out;                        // [16,100,4]
    float* outScores = out + NB * MAXDET * 4;      // [16,100]
    int*   outLabels = (int*)out + NB * MAXDET * 5;// [16,100] int32 (bit-aliased)
    float* outRot    = out + NB * MAXDET * 6;      // [16,100,3]
    float* outTr     = out + NB * MAXDET * 9;      // [16,100,3]
    const size_t r = (size_t)bI * MAXDET + tid;
    if (hi) {
      const unsigned flat = ~(unsigned)key;        // c*200 + rank
      const int lab = (int)(flat / TOPK);
      const unsigned anchor = ws_anchor[(size_t)bI * FLATN + flat];
      const float4 bb = *(const float4*)(boxes + ((size_t)bI * NANCH + anchor) * 4);
      outBoxes[r * 4 + 0] = bb.x; outBoxes[r * 4 + 1] = bb.y;
      outBoxes[r * 4 + 2] = bb.z; outBoxes[r * 4 + 3] = bb.w;
      outScores[r] = __uint_as_float(hi);
      outLabels[r] = lab;
      const float* rp = rot + ((size_t)bI * NANCH + anchor) * 3;
      const float* tp = tr  + ((size_t)bI * NANCH + anchor) * 3;
      outRot[r * 3 + 0] = rp[0]; outRot[r * 3 + 1] = rp[1]; outRot[r * 3 + 2] = rp[2];
      outTr[r * 3 + 0] = tp[0]; outTr[r * 3 + 1] = tp[1]; outTr[r * 3 + 2] = tp[2];
    } else {
      outBoxes[r * 4 + 0] = -1.f; outBoxes[r * 4 + 1] = -1.f;
      outBoxes[r * 4 + 2] = -1.f; outBoxes[r * 4 + 3] = -1.f;
      outScores[r] = -1.f;
      outLabels[r] = -1;
      outRot[r * 3 + 0] = -1.f; outRot[r * 3 + 1] = -1.f; outRot[r * 3 + 2] = -1.f;
      outTr[r * 3 + 0] = -1.f; outTr[r * 3 + 1] = -1.f; outTr[r * 3 + 2] = -1.f;
    }
  }
}

extern "C" void kernel_launch(void* const* d_in, const int* in_sizes, int n_in,
                              void* d_out, int out_size, void* d_ws, size_t ws_size,
                              hipStream_t stream) {
  const float* boxes = (const float*)d_in[0];
  const float* cls   = (const float*)d_in[1];
  const float* rot   = (const float*)d_in[2];
  const float* tr    = (const float*)d_in[3];
  unsigned int* ws_hi     = (unsigned int*)d_ws;                    // [16,30,200]
  unsigned int* ws_anchor = ws_hi + (size_t)NB * NCLS * TOPK;       // [16,30,200]
  fd42709_topk_nms<<<NB * NCLS, BDIM, 0, stream>>>(boxes, cls, ws_hi, ws_anchor);
  fd42709_final_topk<<<NB, BDIM, 0, stream>>>(boxes, rot, tr, ws_hi, ws_anchor,
                                              (float*)d_out);
}